// ChannelCompression_26027501814565
// MI455X (gfx1250) — compile-verified
//
#include <hip/hip_runtime.h>

// ChannelCompression: rolled (circulant) 1x1 conv over 16 channels + bias,
// leaky_relu(0.1), then per-(batch, out-channel) min-max normalization over
// the 512x512 spatial dims.
//
// Memory-bound: pass1 reads x (256 MiB) -> per-(b,o) min/max only; pass2
// re-reads x (256 MiB), recomputes, writes normalized out (256 MiB).
// ~0.8 GB total -> ~35 us at 23.3 TB/s. The 16x16x16 per-pixel matmul maps
// onto 4 chained V_WMMA_F32_16X16X4_F32 (M=16 out ch, N=16 px, K=16 in ch).
//
// Each wave processes 32 adjacent pixels (two 16-px WMMA tiles) per loop
// iteration, so every 128B cacheline of x and out is consumed entirely by
// one wave in one iteration -> non-temporal (TH=NT) loads/stores are safe
// and keep the zero-reuse 0.8 GB stream out of L2.

#define IN_C 16
#define OUT_C 16
#define SPATIAL (512 * 512)       // 262144 pixels per (b, c)
#define NPAIRS (SPATIAL / 32)     // 8192 32-pixel (2-tile) groups per batch
#define NBATCH 16
#define EPS 1e-8f
#define NEG_SLOPE 0.1f
#define WAVES_PER_BLOCK 8
#define BLOCK_THREADS 256
#define GRID_X 256

typedef __attribute__((ext_vector_type(2))) float v2f;
typedef __attribute__((ext_vector_type(8))) float v8f;

// Monotonic float -> unsigned encoding so unsigned atomicMin/Max order floats.
__device__ __forceinline__ unsigned enc_f(float f) {
    unsigned u = __float_as_uint(f);
    return (u & 0x80000000u) ? ~u : (u | 0x80000000u);
}
__device__ __forceinline__ float dec_f(unsigned e) {
    return __uint_as_float((e & 0x80000000u) ? (e & 0x7FFFFFFFu) : ~e);
}

// One 16-pixel tile of the circulant conv: D(16 out ch x 16 px) =
// sum over 4 chained K=4 WMMAs of A(W2 tile) x B(x tile), C seeded with bias.
//
// f32 16x4 A layout: lane L (M = L&15): VGPR0 = K{0|2}, VGPR1 = K{1|3}
//   (K offset +2 for lanes 16-31). B (4x16) analogous: VGPR0 holds row
//   k0+2*half across lanes (N = lane&15), VGPR1 holds row k0+2*half+1.
// C/D: VGPR r, lanes 0-15 -> (M=r, N=lane), lanes 16-31 -> (M=r+8, N=lane-16).
__device__ __forceinline__ v8f conv_tile(const float* __restrict__ xb,
                                         const v2f* __restrict__ A,
                                         int p, int lane, float bias0) {
    const int n  = lane & 15;
    const int h2 = (lane >> 4) << 1;   // 0 for lanes 0-15, 2 for lanes 16-31
    v8f acc;
#pragma unroll
    for (int i = 0; i < 8; ++i) acc[i] = bias0;
#pragma unroll
    for (int j = 0; j < 4; ++j) {
        const int c0 = 4 * j + h2;
        v2f bm;
        bm.x = __builtin_nontemporal_load(&xb[(size_t)c0       * SPATIAL + p + n]);
        bm.y = __builtin_nontemporal_load(&xb[(size_t)(c0 + 1) * SPATIAL + p + n]);
        acc = __builtin_amdgcn_wmma_f32_16x16x4_f32(
            /*neg_a=*/false, A[j], /*neg_b=*/false, bm,
            /*c_mod=*/(short)0, acc, /*reuse_a=*/false, /*reuse_b=*/false);
    }
    return acc;
}

// Build the per-lane A fragments for the 4 K-steps from the 16-entry weight
// vector: W2[o][c] = w[(c - o) & 15].
__device__ __forceinline__ void build_A(const float* __restrict__ w, int lane,
                                        v2f* A) {
    const int o  = lane & 15;
    const int h2 = (lane >> 4) << 1;
#pragma unroll
    for (int j = 0; j < 4; ++j) {
        const int c0 = 4 * j + h2;
        A[j].x = w[(c0     - o) & 15];
        A[j].y = w[(c0 + 1 - o) & 15];
    }
}

__global__ void cc_init(unsigned* __restrict__ ws_min,
                        unsigned* __restrict__ ws_max) {
    const int i = threadIdx.x;           // 256 threads cover all (b, ch)
    ws_min[i] = 0xFFFFFFFFu;             // encoded +inf identity
    ws_max[i] = 0x00000000u;             // encoded -inf identity
}

__global__ __launch_bounds__(BLOCK_THREADS)
void cc_minmax(const float* __restrict__ x, const float* __restrict__ w,
               const float* __restrict__ bias,
               unsigned* __restrict__ ws_min, unsigned* __restrict__ ws_max) {
    const int b    = blockIdx.y;
    const int lane = threadIdx.x & 31;
    const int wave = threadIdx.x >> 5;
    const int half = lane >> 4;
    const float b0 = bias[0];

    v2f A[4];
    build_A(w, lane, A);
    const float* xb = x + (size_t)b * IN_C * SPATIAL;

    float vmin[8], vmax[8];
#pragma unroll
    for (int r = 0; r < 8; ++r) { vmin[r] = 3.4e38f; vmax[r] = -3.4e38f; }

    // Uniform per-wave trip count: no EXEC divergence around WMMA.
    for (int q = blockIdx.x * WAVES_PER_BLOCK + wave; q < NPAIRS;
         q += gridDim.x * WAVES_PER_BLOCK) {
        const int p = q * 32;   // two adjacent 16-px tiles = full 128B lines
        v8f c0 = conv_tile(xb, A, p,      lane, b0);
        v8f c1 = conv_tile(xb, A, p + 16, lane, b0);
#pragma unroll
        for (int r = 0; r < 8; ++r) {
            float y0 = c0[r];
            float y1 = c1[r];
            y0 = fmaxf(y0, NEG_SLOPE * y0);   // leaky_relu(0.1)
            y1 = fmaxf(y1, NEG_SLOPE * y1);
            vmin[r] = fminf(vmin[r], fminf(y0, y1));
            vmax[r] = fmaxf(vmax[r], fmaxf(y0, y1));
        }
    }

    // Reduce across the 16 lanes of each half (masks 1..8 stay in-half).
#pragma unroll
    for (int m = 1; m < 16; m <<= 1) {
#pragma unroll
        for (int r = 0; r < 8; ++r) {
            vmin[r] = fminf(vmin[r], __shfl_xor(vmin[r], m, 32));
            vmax[r] = fmaxf(vmax[r], __shfl_xor(vmax[r], m, 32));
        }
    }

    __shared__ float smin[WAVES_PER_BLOCK][16];
    __shared__ float smax[WAVES_PER_BLOCK][16];
    if ((lane & 15) == 0) {                 // lanes 0 and 16 of each wave
#pragma unroll
        for (int r = 0; r < 8; ++r) {
            smin[wave][half * 8 + r] = vmin[r];
            smax[wave][half * 8 + r] = vmax[r];
        }
    }
    __syncthreads();

    if (threadIdx.x < 16) {
        const int ch = threadIdx.x;
        float mn = smin[0][ch], mx = smax[0][ch];
#pragma unroll
        for (int wv = 1; wv < WAVES_PER_BLOCK; ++wv) {
            mn = fminf(mn, smin[wv][ch]);
            mx = fmaxf(mx, smax[wv][ch]);
        }
        atomicMin(&ws_min[b * 16 + ch], enc_f(mn));
        atomicMax(&ws_max[b * 16 + ch], enc_f(mx));
    }
}

__global__ __launch_bounds__(BLOCK_THREADS)
void cc_normalize(const float* __restrict__ x, const float* __restrict__ w,
                  const float* __restrict__ bias,
                  const unsigned* __restrict__ ws_min,
                  const unsigned* __restrict__ ws_max,
                  float* __restrict__ out) {
    const int b    = blockIdx.y;
    const int lane = threadIdx.x & 31;
    const int wave = threadIdx.x >> 5;
    const int half = lane >> 4;
    const int n    = lane & 15;
    const float b0 = bias[0];

    v2f A[4];
    build_A(w, lane, A);
    const float* xb = x + (size_t)b * IN_C * SPATIAL;
    float* ob       = out + (size_t)b * OUT_C * SPATIAL;

    float mn[8], inv[8];
#pragma unroll
    for (int r = 0; r < 8; ++r) {
        const int ch = half * 8 + r;       // matches C/D row mapping
        const float lo = dec_f(ws_min[b * 16 + ch]);
        const float hi = dec_f(ws_max[b * 16 + ch]);
        mn[r]  = lo;
        inv[r] = 1.0f / (hi - lo + EPS);
    }

    for (int q = blockIdx.x * WAVES_PER_BLOCK + wave; q < NPAIRS;
         q += gridDim.x * WAVES_PER_BLOCK) {
        const int p = q * 32;
        v8f c0 = conv_tile(xb, A, p,      lane, b0);
        v8f c1 = conv_tile(xb, A, p + 16, lane, b0);
#pragma unroll
        for (int r = 0; r < 8; ++r) {
            float y0 = c0[r];
            float y1 = c1[r];
            y0 = fmaxf(y0, NEG_SLOPE * y0);
            y1 = fmaxf(y1, NEG_SLOPE * y1);
            // Per r: lanes 0-15 write channel r, lanes 16-31 channel r+8;
            // the two sub-tiles together cover a full 128B line per channel.
            float* rowp = ob + (size_t)(half * 8 + r) * SPATIAL + p + n;
            __builtin_nontemporal_store((y0 - mn[r]) * inv[r], rowp);
            __builtin_nontemporal_store((y1 - mn[r]) * inv[r], rowp + 16);
        }
    }
}

extern "C" void kernel_launch(void* const* d_in, const int* in_sizes, int n_in,
                              void* d_out, int out_size, void* d_ws, size_t ws_size,
                              hipStream_t stream) {
    const float* x    = (const float*)d_in[0];   // [16,16,512,512] fp32
    const float* w    = (const float*)d_in[1];   // [16] fp32
    const float* bias = (const float*)d_in[2];   // [1] fp32
    float* out        = (float*)d_out;           // [16,16,512,512] fp32

    unsigned* ws_min = (unsigned*)d_ws;          // 256 encoded floats
    unsigned* ws_max = ws_min + NBATCH * OUT_C;  // 256 encoded floats

    cc_init<<<1, 256, 0, stream>>>(ws_min, ws_max);

    dim3 grid(GRID_X, NBATCH);
    cc_minmax<<<grid, BLOCK_THREADS, 0, stream>>>(x, w, bias, ws_min, ws_max);
    cc_normalize<<<grid, BLOCK_THREADS, 0, stream>>>(x, w, bias, ws_min, ws_max,
                                                     out);
}